// SelfAttention_39024072851990
// MI455X (gfx1250) — compile-verified
//
#include <hip/hip_runtime.h>

#define CC  256
#define CQK 32
#define NN  4096
#define BB  4
#define VPAD 20   // V LDS row stride in dwords (16B-aligned rows, conflict-free b64 reads)

typedef __attribute__((ext_vector_type(2))) float v2f;
typedef __attribute__((ext_vector_type(4))) float v4f;
typedef __attribute__((ext_vector_type(8))) float v8f;

__device__ __forceinline__ v8f wmma_f32(v2f a, v2f b, v8f c) {
  // V_WMMA_F32_16X16X4_F32 : D(16x16 f32) = A(16x4 f32) * B(4x16 f32) + C
  return __builtin_amdgcn_wmma_f32_16x16x4_f32(
      /*neg_a=*/false, a, /*neg_b=*/false, b,
      /*c_mod=*/(short)0, c, /*reuse_a=*/false, /*reuse_b=*/false);
}

// gfx1250 async DMA: global -> LDS, tracked by ASYNCcnt (ISA §10 / §15.18.3).
// VDST operand = per-lane LDS byte offset, VADDR = 64-bit global address.
__device__ __forceinline__ void async_g2l_b64(unsigned lds_off, const void* g) {
  asm volatile("global_load_async_to_lds_b64 %0, %1, off"
               :: "v"(lds_off), "v"(g) : "memory");
}
__device__ __forceinline__ void async_g2l_b128(unsigned lds_off, const void* g) {
  asm volatile("global_load_async_to_lds_b128 %0, %1, off"
               :: "v"(lds_off), "v"(g) : "memory");
}
__device__ __forceinline__ void wait_async0() {
  asm volatile("s_wait_asynccnt 0x0" ::: "memory");
}

// ---------------------------------------------------------------------------
// Kernel 1: fused 1x1-conv projections (unchanged from round 1; WMMA-clean).
//   otile 0..1  -> Wq -> QT[b][n][32] (transposed), 2..3 -> Wk -> Kb[b][32][N],
//   4..19 -> Wv -> Vb[b][256][N].  One wave = one 16x16 tile, 64 f32 WMMAs.
// ---------------------------------------------------------------------------
__global__ void proj_kernel(const float* __restrict__ x,
                            const float* __restrict__ Wq, const float* __restrict__ bq,
                            const float* __restrict__ Wk, const float* __restrict__ bk,
                            const float* __restrict__ Wv, const float* __restrict__ bv,
                            float* __restrict__ QT, float* __restrict__ Kb,
                            float* __restrict__ Vb) {
  const int lane  = threadIdx.x & 31;
  const int wid   = blockIdx.x * (blockDim.x >> 5) + (threadIdx.x >> 5);
  const int NT    = NN / 16;
  int t = wid;
  const int ntile = t % NT;  t /= NT;
  const int otile = t % 20;  t /= 20;
  const int b     = t;
  if (b >= BB) return;

  const int n0    = ntile * 16;
  const int j     = lane & 15;
  const int half  = lane >> 4;
  const int khalf = half * 2;

  const float* Wrow; const float* bias; int o0;
  if (otile < 2)      { Wrow = Wq; bias = bq; o0 = otile * 16; }
  else if (otile < 4) { Wrow = Wk; bias = bk; o0 = (otile - 2) * 16; }
  else                { Wrow = Wv; bias = bv; o0 = (otile - 4) * 16; }

  const float* xb = x + (size_t)b * CC * NN;

  v8f acc = {};
  #pragma unroll 8
  for (int cb = 0; cb < CC; cb += 4) {
    v2f a = *(const v2f*)&Wrow[(o0 + j) * CC + cb + khalf];
    v2f bf;
    bf.x = xb[(size_t)(cb + khalf + 0) * NN + n0 + j];
    bf.y = xb[(size_t)(cb + khalf + 1) * NN + n0 + j];
    acc = wmma_f32(a, bf, acc);
  }

  float d[8];
  #pragma unroll
  for (int v = 0; v < 8; ++v) d[v] = acc[v] + bias[o0 + v + half * 8];

  if (otile < 2) {
    float* dst = QT + ((size_t)b * NN + (n0 + j)) * CQK + o0 + half * 8;
    v4f s0 = {d[0], d[1], d[2], d[3]};
    v4f s1 = {d[4], d[5], d[6], d[7]};
    *(v4f*)(dst + 0) = s0;
    *(v4f*)(dst + 4) = s1;
  } else if (otile < 4) {
    float* dst = Kb + (size_t)b * CQK * NN;
    #pragma unroll
    for (int v = 0; v < 8; ++v)
      dst[(size_t)(o0 + v + half * 8) * NN + n0 + j] = d[v];
  } else {
    float* dst = Vb + (size_t)b * CC * NN;
    #pragma unroll
    for (int v = 0; v < 8; ++v)
      dst[(size_t)(o0 + v + half * 8) * NN + n0 + j] = d[v];
  }
}

// ---------------------------------------------------------------------------
// Kernel 2: flash attention, LDS-staged K/V with async global->LDS DMA.
// Block = 256 threads = 8 waves = 4 query blocks x 2 channel halves (same b).
// Per 16-key step the block shares one K tile (32x16, 2KB) and one V tile
// (256x16, padded rows, 20KB) staged by GLOBAL_LOAD_ASYNC_TO_LDS, double-
// buffered so the DMA overlaps the 40 f32 WMMAs per wave per step.
// ---------------------------------------------------------------------------
__global__ void attn_kernel(const float* __restrict__ x,
                            const float* __restrict__ QT,
                            const float* __restrict__ Kmat,
                            const float* __restrict__ Vmat,
                            const float* __restrict__ gamma,
                            float* __restrict__ out) {
  __shared__ float kt[2][CQK * 16];     //  4 KB total
  __shared__ float vt[2][CC * VPAD];    // 40 KB total

  const int tid  = threadIdx.x;
  const int lane = tid & 31;
  const int w    = tid >> 5;            // wave 0..7
  const int ch   = w & 1;
  const int b    = blockIdx.x >> 6;     // 64 nb-groups per batch
  const int nb   = (blockIdx.x & 63) * 4 + (w >> 1);

  const int n0    = nb * 16;
  const int j     = lane & 15;
  const int half  = lane >> 4;
  const int khalf = half * 2;
  const int c0    = ch * 128;

  const float* qt = QT   + (size_t)b * NN * CQK;
  const float* kp = Kmat + (size_t)b * CQK * NN;
  const float* vp = Vmat + (size_t)b * CC * NN;

  const unsigned ktb0 = (unsigned)(uintptr_t)&kt[0][0];
  const unsigned ktb1 = (unsigned)(uintptr_t)&kt[1][0];
  const unsigned vtb0 = (unsigned)(uintptr_t)&vt[0][0];
  const unsigned vtb1 = (unsigned)(uintptr_t)&vt[1][0];

  // Per-thread staging assignment (block-wide cooperative copy):
  //   K: thread -> (d = tid/8, m = 2*(tid%8)), one async b64   (32x16 tile)
  //   V: thread -> row c = tid, four async b128                (256x16 tile)
  const int kd = tid >> 3;
  const int km = (tid & 7) << 1;
  const unsigned k_ldsoff = (unsigned)((kd * 16 + km) * 4);
  const unsigned v_ldsoff = (unsigned)(tid * VPAD * 4);

  // Preload Q B-fragments: B[k=d][j=n] = QT[n0+j][d], contiguous float2.
  v2f qf[8];
  #pragma unroll
  for (int c = 0; c < 8; ++c)
    qf[c] = *(const v2f*)&qt[(n0 + j) * CQK + c * 4 + khalf];

  v8f acc[8];
  #pragma unroll
  for (int tt = 0; tt < 8; ++tt) acc[tt] = (v8f){};

  float runmax = -__builtin_inff();
  float runsum = 0.0f;

  // Prologue: stage tile 0 into buffer 0.
  {
    async_g2l_b64(ktb0 + k_ldsoff, kp + (size_t)kd * NN + km);
    const float* vrow = vp + (size_t)tid * NN;
    #pragma unroll
    for (int k = 0; k < 4; ++k)
      async_g2l_b128(vtb0 + v_ldsoff + k * 16, vrow + k * 4);
  }

  const int NITER = NN / 16;
  for (int it = 0; it < NITER; ++it) {
    const int mb  = it * 16;
    const int cur = it & 1;

    wait_async0();        // own async copies for the current buffer done
    __syncthreads();      // everyone's copies visible; prev reads finished

    if (it + 1 < NITER) { // kick DMA for the next tile while we compute
      const unsigned kb2 = cur ? ktb0 : ktb1;
      const unsigned vb2 = cur ? vtb0 : vtb1;
      async_g2l_b64(kb2 + k_ldsoff, kp + (size_t)kd * NN + (mb + 16) + km);
      const float* vrow = vp + (size_t)tid * NN + (mb + 16);
      #pragma unroll
      for (int k = 0; k < 4; ++k)
        async_g2l_b128(vb2 + v_ldsoff + k * 16, vrow + k * 4);
    }

    const float* ktc = &kt[cur][0];
    const float* vtc = &vt[cur][0];

    // --- S^T tile: rows m = mb + v + half*8 (VGPRs), cols n = n0+j (lanes)
    v8f st = (v8f){};
    #pragma unroll
    for (int c = 0; c < 8; ++c) {
      v2f a;                               // A[i=m][k=d] = K[d][m] from LDS
      a.x = ktc[(c * 4 + khalf + 0) * 16 + j];
      a.y = ktc[(c * 4 + khalf + 1) * 16 + j];
      st = wmma_f32(a, qf[c], st);
    }

    // --- online softmax over keys (fixed query n = lane column)
    float tmax = st[0];
    #pragma unroll
    for (int v = 1; v < 8; ++v) tmax = fmaxf(tmax, st[v]);
    tmax = fmaxf(tmax, __shfl_xor(tmax, 16, 32));
    const float newmax = fmaxf(runmax, tmax);
    const float scale  = __expf(runmax - newmax);

    float p[8];
    float psum = 0.0f;
    #pragma unroll
    for (int v = 0; v < 8; ++v) { p[v] = __expf(st[v] - newmax); psum += p[v]; }
    psum  += __shfl_xor(psum, 16, 32);
    runsum = runsum * scale + psum;
    runmax = newmax;

    if (scale < 1.0f) {                    // VALU-only divergence: safe
      #pragma unroll
      for (int tt = 0; tt < 8; ++tt)
        #pragma unroll
        for (int v = 0; v < 8; ++v) acc[tt][v] *= scale;
    }

    // --- re-layout P (D layout) into B fragments via lane shuffles.
    v2f pf[4];
    #pragma unroll
    for (int cch = 0; cch < 4; ++cch) {
      const int r0 = cch * 4;
      float a0 = __shfl(p[(r0 + 0) & 7], (((r0 + 0) >> 3) << 4) + j, 32);
      float a2 = __shfl(p[(r0 + 2) & 7], (((r0 + 2) >> 3) << 4) + j, 32);
      float a1 = __shfl(p[(r0 + 1) & 7], (((r0 + 1) >> 3) << 4) + j, 32);
      float a3 = __shfl(p[(r0 + 3) & 7], (((r0 + 3) >> 3) << 4) + j, 32);
      pf[cch].x = half ? a2 : a0;
      pf[cch].y = half ? a3 : a1;
    }

    // --- O(16c x 16n) += V(16c x 16m) * P^T(16m x 16n): 8 tiles x 4 chunks
    #pragma unroll
    for (int tt = 0; tt < 8; ++tt) {
      const float* vrow = &vtc[(c0 + tt * 16 + j) * VPAD];
      #pragma unroll
      for (int cch = 0; cch < 4; ++cch) {
        v2f a = *(const v2f*)&vrow[cch * 4 + khalf];   // ds_load_b64
        acc[tt] = wmma_f32(a, pf[cch], acc[tt]);
      }
    }
  }

  // --- epilogue: out = gamma * O / L + x
  const float g   = gamma[0];
  const float inv = 1.0f / runsum;
  #pragma unroll
  for (int tt = 0; tt < 8; ++tt) {
    #pragma unroll
    for (int v = 0; v < 8; ++v) {
      const int c = c0 + tt * 16 + v + half * 8;
      const size_t idx = ((size_t)b * CC + c) * NN + n0 + j;
      out[idx] = g * acc[tt][v] * inv + x[idx];
    }
  }
}

extern "C" void kernel_launch(void* const* d_in, const int* in_sizes, int n_in,
                              void* d_out, int out_size, void* d_ws, size_t ws_size,
                              hipStream_t stream) {
  const float* x     = (const float*)d_in[0];
  const float* Wq    = (const float*)d_in[1];
  const float* bq    = (const float*)d_in[2];
  const float* Wk    = (const float*)d_in[3];
  const float* bk    = (const float*)d_in[4];
  const float* Wv    = (const float*)d_in[5];
  const float* bv    = (const float*)d_in[6];
  const float* gamma = (const float*)d_in[7];
  float* out = (float*)d_out;

  float* ws = (float*)d_ws;
  float* QT = ws;                                   // [B][N][32]   2 MB
  float* Kb = QT + (size_t)BB * NN * CQK;           // [B][32][N]   2 MB
  float* Vb = Kb + (size_t)BB * CQK * NN;           // [B][256][N] 16 MB

  const int waves1 = BB * 20 * (NN / 16);           // 20480 tiles
  proj_kernel<<<waves1 / 8, 256, 0, stream>>>(x, Wq, bq, Wk, bk, Wv, bv, QT, Kb, Vb);

  const int blocks2 = BB * (NN / 16) / 4;           // 256 blocks x 8 waves
  attn_kernel<<<blocks2, 256, 0, stream>>>(x, QT, Kb, Vb, gamma, out);
}